// BiMambaBackbone_62216896250087
// MI455X (gfx1250) — compile-verified
//
#include <hip/hip_runtime.h>
#include <math.h>

#define D_MODEL 768
#define D_STATE 16
#define NHEADS  24
#define HEADDIM 64
#define D_INNER 1536
#define CONV_DIM 1568
#define D_IN_PROJ 3128
#define SEQ 2048
#define BATCH 2
#define NTOK (BATCH*SEQ)

typedef __attribute__((ext_vector_type(2))) float v2f;
typedef __attribute__((ext_vector_type(8))) float v8f;

__device__ __forceinline__ float sigmoidf_(float x){ return 1.0f/(1.0f+expf(-x)); }

// ---------------------------------------------------------------------------
// GEMM: C[M,N] = A[M,K] @ B[K,N]; mode 0: C = acc (+ep[col] bias if ep)
//                                 mode 1: C[r][c] += acc * ep[c]   (residual)
// block = 256 threads (8 wave32), block tile 128x64, wave tile 32x32
// inner: V_WMMA_F32_16X16X4_F32, A/B staged via LDS (padded strides).
// M must be a multiple of 128, K a multiple of 32; N arbitrary (guarded).
// ---------------------------------------------------------------------------
__global__ __launch_bounds__(256) void gemm_wmma_f32(
    const float* __restrict__ A, int lda,
    const float* __restrict__ B, int ldb,
    float* __restrict__ C, int ldc,
    int N, int K,
    const float* __restrict__ ep, int mode)
{
  __shared__ float sA[128*36];   // 128 rows x 32 cols, stride 36 (pad)
  __shared__ float sB[32*68];    // 32 rows x 64 cols, stride 68 (pad)
  const int tid  = threadIdx.x;
  const int lane = tid & 31;
  const int wid  = tid >> 5;
  const int m0 = blockIdx.y * 128;
  const int n0 = blockIdx.x * 64;
  const int wm = (wid >> 1) * 32;   // 4x2 wave grid
  const int wn = (wid & 1) * 32;
  const int hi = lane >> 4;         // 0: K pair {0,1}; 1: K pair {2,3}
  const int lw = lane & 15;

  v8f acc[2][2] = {};

  for (int k0 = 0; k0 < K; k0 += 32) {
    if (k0 + 32 < K) {              // prefetch next chunk -> global_prefetch
      __builtin_prefetch(&A[(size_t)(m0 + (tid & 127)) * lda + k0 + 32], 0, 1);
      __builtin_prefetch(&B[(size_t)(k0 + 32 + (tid & 31)) * ldb + n0], 0, 1);
    }
    // stage A: 128x32 (always in-bounds: M%128==0, K%32==0)
    #pragma unroll
    for (int i = 0; i < 4; ++i) {
      int idx = tid + i*256;                 // 128 rows x 8 float4
      int row = idx >> 3, c4 = (idx & 7) << 2;
      float4 v = *(const float4*)&A[(size_t)(m0+row)*lda + k0 + c4];
      *(float4*)&sA[row*36 + c4] = v;
    }
    // stage B: 32x64 (guard columns against N)
    #pragma unroll
    for (int i = 0; i < 2; ++i) {
      int idx = tid + i*256;                 // 32 rows x 16 float4
      int row = idx >> 4, c4 = (idx & 15) << 2;
      int gc = n0 + c4;
      const float* brow = &B[(size_t)(k0+row)*ldb];
      float4 v;
      if (gc + 3 < N) v = *(const float4*)&brow[gc];
      else {
        v.x = (gc+0 < N) ? brow[gc+0] : 0.f;
        v.y = (gc+1 < N) ? brow[gc+1] : 0.f;
        v.z = (gc+2 < N) ? brow[gc+2] : 0.f;
        v.w = (gc+3 < N) ? brow[gc+3] : 0.f;
      }
      *(float4*)&sB[row*68 + c4] = v;
    }
    __syncthreads();
    #pragma unroll
    for (int kk = 0; kk < 32; kk += 4) {
      v2f a[2], b[2];
      #pragma unroll
      for (int s = 0; s < 2; ++s)          // A frag: M=lw, K={kk+2hi, kk+2hi+1}
        a[s] = *(const v2f*)&sA[(wm + s*16 + lw)*36 + kk + 2*hi];
      #pragma unroll
      for (int s = 0; s < 2; ++s) {        // B frag: N=lw, same K split
        b[s].x = sB[(kk + 2*hi    )*68 + wn + s*16 + lw];
        b[s].y = sB[(kk + 2*hi + 1)*68 + wn + s*16 + lw];
      }
      #pragma unroll
      for (int sm = 0; sm < 2; ++sm)
        #pragma unroll
        for (int sn = 0; sn < 2; ++sn)
          acc[sm][sn] = __builtin_amdgcn_wmma_f32_16x16x4_f32(
              false, a[sm], false, b[sn], (short)0, acc[sm][sn], false, false);
    }
    __syncthreads();
  }
  // D layout: VGPR j -> row = j + 8*hi, col = lw
  #pragma unroll
  for (int sm = 0; sm < 2; ++sm)
    #pragma unroll
    for (int sn = 0; sn < 2; ++sn) {
      int col = n0 + wn + sn*16 + lw;
      if (col >= N) continue;
      #pragma unroll
      for (int j = 0; j < 8; ++j) {
        int row = m0 + wm + sm*16 + j + 8*hi;
        float v = acc[sm][sn][j];
        size_t o = (size_t)row*ldc + col;
        if (mode == 0) { if (ep) v += ep[col]; C[o] = v; }
        else           { C[o] += v * ep[col]; }
      }
    }
}

// ---------------------------------------------------------------------------
// Depthwise causal conv(4) + SiLU + split into x/B/C, plus softplus(dt).
// dir==1 runs the reversed-time conv in place (no flip materialized).
// ---------------------------------------------------------------------------
__global__ __launch_bounds__(256) void conv_split_kernel(
    const float* __restrict__ Z,
    const float* __restrict__ cw, const float* __restrict__ cb,
    const float* __restrict__ dtb,
    float* __restrict__ XH, float* __restrict__ Bc,
    float* __restrict__ Cc, float* __restrict__ DTo, int dir)
{
  int tok = blockIdx.x;
  int b = tok >> 11, t = tok & (SEQ-1);
  const float* zrow = Z + (size_t)tok * D_IN_PROJ;
  for (int c = threadIdx.x; c < CONV_DIM + NHEADS; c += 256) {
    if (c < CONV_DIM) {
      float acc = cb[c];
      #pragma unroll
      for (int k = 0; k < 4; ++k) {
        int tt = dir ? (t + 3 - k) : (t - 3 + k);
        if (tt >= 0 && tt < SEQ)
          acc += cw[c*4 + k] * Z[((size_t)(b*SEQ + tt))*D_IN_PROJ + D_INNER + c];
      }
      acc = acc * sigmoidf_(acc);                      // SiLU
      if (c < D_INNER)                XH[(size_t)tok*D_INNER + c] = acc;
      else if (c < D_INNER + D_STATE) Bc[(size_t)tok*D_STATE + (c - D_INNER)] = acc;
      else                            Cc[(size_t)tok*D_STATE + (c - D_INNER - D_STATE)] = acc;
    } else {
      int h = c - CONV_DIM;
      float v = zrow[D_INNER + CONV_DIM + h] + dtb[h];
      DTo[(size_t)tok*NHEADS + h] = (v > 20.f) ? v : log1pf(expf(v));  // softplus
    }
  }
}

// ---------------------------------------------------------------------------
// Selective scan. One block per (headdim-chunk, head, batch); thread=(p,n).
// State h[p][n] lives in a register; reduce over n via shfl width-16 (wave32).
// ---------------------------------------------------------------------------
__global__ __launch_bounds__(256) void scan_kernel(
    const float* __restrict__ XH, const float* __restrict__ Bc,
    const float* __restrict__ Cc, const float* __restrict__ DTi,
    const float* __restrict__ A_log, const float* __restrict__ Dh,
    float* __restrict__ Y, int dir)
{
  int n = threadIdx.x & 15;
  int p = blockIdx.x*16 + (threadIdx.x >> 4);
  int h = blockIdx.y, b = blockIdx.z;
  float A = -expf(A_log[h]);
  float D = Dh[h];
  float hs = 0.f;
  for (int s = 0; s < SEQ; ++s) {
    int t = dir ? (SEQ-1-s) : s;
    size_t tok = (size_t)(b*SEQ + t);
    float dt = DTi[tok*NHEADS + h];
    float dA = expf(dt * A);
    float xv = XH[tok*D_INNER + h*HEADDIM + p];
    float Bv = Bc[tok*D_STATE + n];
    hs = hs * dA + dt * xv * Bv;
    float part = hs * Cc[tok*D_STATE + n];
    #pragma unroll
    for (int o = 8; o >= 1; o >>= 1) part += __shfl_xor(part, o, 16);
    if (n == 0) Y[tok*D_INNER + h*HEADDIM + p] = part + xv * D;
  }
}

// ---------------------------------------------------------------------------
// g = y * silu(z); g *= rsqrt(mean(g^2)+1e-5); y = g * norm_w  (in place on Y)
// ---------------------------------------------------------------------------
__global__ __launch_bounds__(256) void gate_rms_kernel(
    float* __restrict__ Y, const float* __restrict__ Z,
    const float* __restrict__ nw)
{
  __shared__ float red[8];
  int tok = blockIdx.x;
  float g[6]; float ss = 0.f;
  #pragma unroll
  for (int i = 0; i < 6; ++i) {
    int c = threadIdx.x + i*256;
    float y = Y[(size_t)tok*D_INNER + c];
    float z = Z[(size_t)tok*D_IN_PROJ + c];
    float gv = y * (z * sigmoidf_(z));
    g[i] = gv; ss += gv*gv;
  }
  #pragma unroll
  for (int o = 16; o >= 1; o >>= 1) ss += __shfl_xor(ss, o, 32);
  if ((threadIdx.x & 31) == 0) red[threadIdx.x >> 5] = ss;
  __syncthreads();
  if (threadIdx.x == 0) {
    float tot = 0.f;
    #pragma unroll
    for (int i = 0; i < 8; ++i) tot += red[i];
    red[0] = rsqrtf(tot / (float)D_INNER + 1e-5f);
  }
  __syncthreads();
  float r = red[0];
  #pragma unroll
  for (int i = 0; i < 6; ++i) {
    int c = threadIdx.x + i*256;
    Y[(size_t)tok*D_INNER + c] = g[i] * r * nw[c];
  }
}

// ---------------------------------------------------------------------------
// out = x * rsqrt(mean(x^2)+1e-6) * (1 + fin_w)
// ---------------------------------------------------------------------------
__global__ __launch_bounds__(256) void final_norm_kernel(
    const float* __restrict__ X, const float* __restrict__ fw,
    float* __restrict__ out)
{
  __shared__ float red[8];
  int tok = blockIdx.x;
  float v[3]; float ss = 0.f;
  #pragma unroll
  for (int i = 0; i < 3; ++i) {
    int c = threadIdx.x + i*256;
    v[i] = X[(size_t)tok*D_MODEL + c];
    ss += v[i]*v[i];
  }
  #pragma unroll
  for (int o = 16; o >= 1; o >>= 1) ss += __shfl_xor(ss, o, 32);
  if ((threadIdx.x & 31) == 0) red[threadIdx.x >> 5] = ss;
  __syncthreads();
  if (threadIdx.x == 0) {
    float tot = 0.f;
    #pragma unroll
    for (int i = 0; i < 8; ++i) tot += red[i];
    red[0] = rsqrtf(tot / (float)D_MODEL + 1e-6f);
  }
  __syncthreads();
  float r = red[0];
  #pragma unroll
  for (int i = 0; i < 3; ++i) {
    int c = threadIdx.x + i*256;
    out[(size_t)tok*D_MODEL + c] = v[i] * r * (1.f + fw[c]);
  }
}

// ---------------------------------------------------------------------------
extern "C" void kernel_launch(void* const* d_in, const int* in_sizes, int n_in,
                              void* d_out, int out_size, void* d_ws, size_t ws_size,
                              hipStream_t stream) {
  const float* pe      = (const float*)d_in[0];
  const float* Wp      = (const float*)d_in[1];
  const float* bp      = (const float*)d_in[2];
  const float* in_w    = (const float*)d_in[3];
  const float* conv_w  = (const float*)d_in[4];
  const float* conv_b  = (const float*)d_in[5];
  const float* dt_bias = (const float*)d_in[6];
  const float* A_log   = (const float*)d_in[7];
  const float* Dh      = (const float*)d_in[8];
  const float* norm_w  = (const float*)d_in[9];
  const float* out_w   = (const float*)d_in[10];
  const float* bi_w    = (const float*)d_in[11];
  const float* ls      = (const float*)d_in[12];
  const float* fin_w   = (const float*)d_in[13];
  float* out = (float*)d_out;

  float* X   = (float*)d_ws;                       // 4096 x 768
  float* Z   = X  + (size_t)NTOK*D_MODEL;          // 4096 x 3128
  float* XH  = Z  + (size_t)NTOK*D_IN_PROJ;        // 4096 x 1536
  float* Bc  = XH + (size_t)NTOK*D_INNER;          // 4096 x 16
  float* Cc  = Bc + (size_t)NTOK*D_STATE;          // 4096 x 16
  float* DT  = Cc + (size_t)NTOK*D_STATE;          // 4096 x 24
  float* Y   = DT + (size_t)NTOK*NHEADS;           // 4096 x 1536
  float* CAT = Y  + (size_t)NTOK*D_INNER;          // 4096 x 1536  [h_f | h_b]

  dim3 blk(256);
  // x = pe @ Wp + bp
  gemm_wmma_f32<<<dim3((D_MODEL+63)/64, NTOK/128), blk, 0, stream>>>(
      pe, 256, Wp, D_MODEL, X, D_MODEL, D_MODEL, 256, bp, 0);

  for (int layer = 0; layer < 2; ++layer) {
    for (int dir = 0; dir < 2; ++dir) {
      int li = layer*2 + dir;
      // zxbcdt = x @ in_w
      gemm_wmma_f32<<<dim3((D_IN_PROJ+63)/64, NTOK/128), blk, 0, stream>>>(
          X, D_MODEL, in_w + (size_t)li*D_MODEL*D_IN_PROJ, D_IN_PROJ,
          Z, D_IN_PROJ, D_IN_PROJ, D_MODEL, nullptr, 0);
      conv_split_kernel<<<NTOK, blk, 0, stream>>>(
          Z, conv_w + (size_t)li*CONV_DIM*4, conv_b + (size_t)li*CONV_DIM,
          dt_bias + (size_t)li*NHEADS, XH, Bc, Cc, DT, dir);
      scan_kernel<<<dim3(HEADDIM/16, NHEADS, BATCH), blk, 0, stream>>>(
          XH, Bc, Cc, DT, A_log + (size_t)li*NHEADS, Dh + (size_t)li*NHEADS, Y, dir);
      gate_rms_kernel<<<NTOK, blk, 0, stream>>>(Y, Z, norm_w + (size_t)li*D_INNER);
      // h_dir = g @ out_w   (written into its half of CAT)
      gemm_wmma_f32<<<dim3((D_MODEL+63)/64, NTOK/128), blk, 0, stream>>>(
          Y, D_INNER, out_w + (size_t)li*D_INNER*D_MODEL, D_MODEL,
          CAT + dir*D_MODEL, 2*D_MODEL, D_MODEL, D_INNER, nullptr, 0);
    }
    // x += ([h_f|h_b] @ bi_w) * ls
    gemm_wmma_f32<<<dim3((D_MODEL+63)/64, NTOK/128), blk, 0, stream>>>(
        CAT, 2*D_MODEL, bi_w + (size_t)layer*2*D_MODEL*D_MODEL, D_MODEL,
        X, D_MODEL, D_MODEL, 2*D_MODEL, ls + (size_t)layer*D_MODEL, 1);
  }
  final_norm_kernel<<<NTOK, blk, 0, stream>>>(X, fin_w, out);
}